// CayleyConv_50019189129917
// MI455X (gfx1250) — compile-verified
//
#include <hip/hip_runtime.h>
#include <hip/hip_bf16.h>
#include <cstdint>

typedef __attribute__((ext_vector_type(2))) float v2f;
typedef __attribute__((ext_vector_type(8))) float v8f;
typedef unsigned int u32x4 __attribute__((ext_vector_type(4)));
typedef int i32x4 __attribute__((ext_vector_type(4)));
typedef int i32x8 __attribute__((ext_vector_type(8)));

static constexpr int B_  = 8;
static constexpr int C_  = 256;          // GEMM K
static constexpr int CO_ = 256;          // GEMM N
static constexpr int H_  = 28;
static constexpr int W_  = 28;
static constexpr int M_B = H_ * W_ * 9;  // 7056 rows / batch = 441 * 16
static constexpr int MTILES = M_B / 16;  // 441
static constexpr int KC  = 16;           // K chunk per LDS stage
static constexpr int NC  = C_ / KC;      // 16 chunks
static constexpr int BP  = 272;          // Bs pitch (dwords): 2*272 % 64 == 32 -> conflict-free
static constexpr int AS_DW = KC * 16;    // 256 dwords per A buffer
static constexpr int BS_DW = KC * BP;    // 4352 dwords per B buffer
static constexpr int LDS_DW = 2 * AS_DW + 2 * BS_DW;   // 9216 dwords = 36864 B

#define EPSF 1e-07f

#if defined(__gfx1250__) && __has_builtin(__builtin_amdgcn_tensor_load_to_lds) && \
    __has_builtin(__builtin_amdgcn_s_wait_tensorcnt)
#define USE_TDM 1
#else
#define USE_TDM 0
#endif

// ---------------- Stage 0: squared channel-mixing weights ----------------
__global__ void sq_kernel(const float* __restrict__ g, float* __restrict__ csq) {
  int i = blockIdx.x * blockDim.x + threadIdx.x;
  float v = g[24 + i];
  csq[i] = v * v;
}

// ---------------- Stage 1: Cayley matrices G[9][3][3] ----------------
__global__ void cayley_kernel(const float* __restrict__ g, float* __restrict__ G) {
  int k = threadIdx.x;
  if (k >= 9) return;
  if (k == 8) {                      // identity goes to slot 4
    float* o = G + 4 * 9;
    o[0] = 1.f; o[1] = 0.f; o[2] = 0.f;
    o[3] = 0.f; o[4] = 1.f; o[5] = 0.f;
    o[6] = 0.f; o[7] = 0.f; o[8] = 1.f;
    return;
  }
  float a = g[3 * k + 0], b = g[3 * k + 1], c = g[3 * k + 2];
  float f00 = 1.f, f01 = -a, f02 = -b;
  float f10 = a,   f11 = 1.f, f12 = -c;
  float f20 = b,   f21 = c,   f22 = 1.f;
  float det = f00 * (f11 * f22 - f12 * f21)
            - f01 * (f10 * f22 - f12 * f20)
            + f02 * (f10 * f21 - f11 * f20);
  float inv = 1.0f / (det + EPSF);
  float i00 = (f11 * f22 - f12 * f21) * inv, i01 = (f02 * f21 - f01 * f22) * inv, i02 = (f01 * f12 - f02 * f11) * inv;
  float i10 = (f12 * f20 - f10 * f22) * inv, i11 = (f00 * f22 - f02 * f20) * inv, i12 = (f02 * f10 - f00 * f12) * inv;
  float i20 = (f10 * f21 - f11 * f20) * inv, i21 = (f01 * f20 - f00 * f21) * inv, i22 = (f00 * f11 - f01 * f10) * inv;
  float s00 = 1.f, s01 = a,   s02 = b;
  float s10 = -a,  s11 = 1.f, s12 = c;
  float s20 = -b,  s21 = -c,  s22 = 1.f;
  int pos = (k < 4) ? k : ((k == 4) ? 8 : k);
  float* o = G + pos * 9;
  o[0] = i00 * s00 + i01 * s10 + i02 * s20;
  o[1] = i00 * s01 + i01 * s11 + i02 * s21;
  o[2] = i00 * s02 + i01 * s12 + i02 * s22;
  o[3] = i10 * s00 + i11 * s10 + i12 * s20;
  o[4] = i10 * s01 + i11 * s11 + i12 * s21;
  o[5] = i10 * s02 + i11 * s12 + i12 * s22;
  o[6] = i20 * s00 + i21 * s10 + i22 * s20;
  o[7] = i20 * s01 + i21 * s11 + i22 * s21;
  o[8] = i20 * s02 + i21 * s12 + i22 * s22;
}

#if USE_TDM
// Issue one 2D TDM tile load: global (strided) -> LDS (optionally padded).
__device__ __forceinline__ void tdm_issue(unsigned ldsByteOff, unsigned long long gaddr,
                                          int w0, int w1, int w2, int w3, int w4, int w5) {
  u32x4 g0;
  g0[0] = 1u;                                            // count=1, user descriptor
  g0[1] = ldsByteOff;                                    // lds_addr (bytes)
  g0[2] = (unsigned)(gaddr & 0xFFFFFFFFull);             // global_addr[31:0]
  g0[3] = (unsigned)((gaddr >> 32) & 0x01FFFFFFull) | (2u << 30);  // addr[56:32] | type=2
  i32x8 g1;
  g1[0] = w0;  // workgroup_mask=0 | data_size | pad ctrl
  g1[1] = w1;  // tensor_dim0[15:0] << 16
  g1[2] = w2;  // tensor_dim0[31:16] | tensor_dim1[15:0] << 16
  g1[3] = w3;  // tensor_dim1[31:16] | tile_dim0 << 16
  g1[4] = w4;  // tile_dim1 | tile_dim2 << 16
  g1[5] = w5;  // tensor_dim0_stride[31:0]
  g1[6] = 0;   // stride0[47:32] | stride1[15:0]
  g1[7] = 0;
  i32x4 z4 = {0, 0, 0, 0};
  i32x8 z8 = {0, 0, 0, 0, 0, 0, 0, 0};
  __builtin_amdgcn_tensor_load_to_lds(g0, g1, z4, z4, z8, 0);
}
#endif

// ---------------- Stage 2: channel-mix GEMM via WMMA f32 16x16x4 ----------------
// Xmix[(b*7056 + r)*256 + o] = sum_c x[b,c,r] * csq[c*256 + o]
// Block: 16 rows (M) x 256 cols (N); 8 waves, each wave owns N subtiles 32*wv and 32*wv+16.
__global__ __launch_bounds__(256) void mix_gemm_kernel(const float* __restrict__ x,
                                                       const float* __restrict__ csq,
                                                       float* __restrict__ Xmix) {
  extern __shared__ float smem[];   // [As ping|As pong|Bs ping|Bs pong]

  int bt = blockIdx.x;
  int mt = bt % MTILES;
  int b  = bt / MTILES;
  int r0 = mt * 16;

  int tid  = threadIdx.x;
  int lane = tid & 31;
  int wv   = tid >> 5;
  int lh   = lane >> 4;
  int ll   = lane & 15;

  const float* xb = x + (size_t)b * C_ * M_B;

  v8f acc0 = {};
  v8f acc1 = {};

#if USE_TDM
  // Double-buffered TDM pipeline: chunk c+1 copied by the Tensor Data Mover
  // while waves run WMMA on chunk c.
  auto issue_chunk = [&](int c0, int buf) {
    if (wv == 0) {
      unsigned asOff = (unsigned)(buf * AS_DW * 4);
      unsigned bsOff = (unsigned)((2 * AS_DW + buf * BS_DW) * 4);
      // A tile: KC rows (stride M_B floats) x 16 floats, packed in LDS (pitch 16)
      tdm_issue(asOff, (unsigned long long)(uintptr_t)(xb + (size_t)c0 * M_B + r0),
                (2 << 16),               // data_size = 4B
                (M_B & 0xFFFF) << 16,    // tensor_dim0 = 7056
                (C_) << 16,              // tensor_dim1 = 256
                (16) << 16,              // tile_dim0 = 16
                KC,                      // tile_dim1 = KC
                M_B);                    // tensor_dim0_stride = 7056
      // B tile: KC rows (stride 256) x 256 floats, LDS pitch 272 via HW padding
      tdm_issue(bsOff, (unsigned long long)(uintptr_t)(csq + (size_t)c0 * CO_),
                (2 << 16) | (1 << 20) | (7 << 22) | (15 << 25),  // 4B, pad 16dw every 256dw
                (CO_) << 16,             // tensor_dim0 = 256
                (C_) << 16,              // tensor_dim1 = 256
                (CO_) << 16,             // tile_dim0 = 256
                KC,                      // tile_dim1 = KC
                CO_);                    // tensor_dim0_stride = 256
    }
  };

  issue_chunk(0, 0);
  for (int ch = 0; ch < NC; ++ch) {
    int buf = ch & 1;
    if (ch + 1 < NC) issue_chunk((ch + 1) * KC, (ch + 1) & 1);
    if (wv == 0) {
      if (ch + 1 < NC) __builtin_amdgcn_s_wait_tensorcnt(2);  // current chunk done
      else             __builtin_amdgcn_s_wait_tensorcnt(0);
    }
    __syncthreads();
    const float* As = smem + buf * AS_DW;
    const float* Bs = smem + 2 * AS_DW + buf * BS_DW;
#pragma unroll
    for (int kk = 0; kk < KC; kk += 4) {
      v2f af, bf0, bf1;
      af[0]  = As[(kk + 2 * lh + 0) * 16 + ll];
      af[1]  = As[(kk + 2 * lh + 1) * 16 + ll];
      bf0[0] = Bs[(kk + 2 * lh + 0) * BP + 32 * wv + ll];
      bf0[1] = Bs[(kk + 2 * lh + 1) * BP + 32 * wv + ll];
      bf1[0] = Bs[(kk + 2 * lh + 0) * BP + 32 * wv + 16 + ll];
      bf1[1] = Bs[(kk + 2 * lh + 1) * BP + 32 * wv + 16 + ll];
      acc0 = __builtin_amdgcn_wmma_f32_16x16x4_f32(false, af, false, bf0,
                                                   (short)0, acc0, false, false);
      acc1 = __builtin_amdgcn_wmma_f32_16x16x4_f32(false, af, false, bf1,
                                                   (short)0, acc1, false, false);
    }
    __syncthreads();
  }
#else
  // Fallback: manual staged copies (single buffer).
  for (int c0 = 0; c0 < C_; c0 += KC) {
    if (tid < 64) {
      int cA = tid >> 2, qA = tid & 3;
      float4 av = *(const float4*)(xb + (size_t)(c0 + cA) * M_B + r0 + 4 * qA);
      *(float4*)(&smem[cA * 16 + 4 * qA]) = av;
    }
#pragma unroll
    for (int i = 0; i < 4; ++i) {
      int idx = tid + (i << 8);
      int rB  = idx >> 6;
      int c4  = idx & 63;
      float4 bv = *(const float4*)(csq + (size_t)(c0 + rB) * CO_ + 4 * c4);
      *(float4*)(&smem[2 * AS_DW + rB * BP + 4 * c4]) = bv;
    }
    __syncthreads();
    const float* As = smem;
    const float* Bs = smem + 2 * AS_DW;
#pragma unroll
    for (int kk = 0; kk < KC; kk += 4) {
      v2f af, bf0, bf1;
      af[0]  = As[(kk + 2 * lh + 0) * 16 + ll];
      af[1]  = As[(kk + 2 * lh + 1) * 16 + ll];
      bf0[0] = Bs[(kk + 2 * lh + 0) * BP + 32 * wv + ll];
      bf0[1] = Bs[(kk + 2 * lh + 1) * BP + 32 * wv + ll];
      bf1[0] = Bs[(kk + 2 * lh + 0) * BP + 32 * wv + 16 + ll];
      bf1[1] = Bs[(kk + 2 * lh + 1) * BP + 32 * wv + 16 + ll];
      acc0 = __builtin_amdgcn_wmma_f32_16x16x4_f32(false, af, false, bf0,
                                                   (short)0, acc0, false, false);
      acc1 = __builtin_amdgcn_wmma_f32_16x16x4_f32(false, af, false, bf1,
                                                   (short)0, acc1, false, false);
    }
    __syncthreads();
  }
#endif

  // D layout: VGPR v, lane: M = v + 8*lh, N = ll (per 16-wide subtile)
  float* dstp = Xmix + ((size_t)b * M_B + r0) * CO_ + 32 * wv + ll;
#pragma unroll
  for (int v = 0; v < 8; ++v) {
    size_t rowOff = (size_t)(v + 8 * lh) * CO_;
    dstp[rowOff]      = acc0[v];
    dstp[rowOff + 16] = acc1[v];
  }
}

// ---------------- Stage 3: Cayley sandwich + transposed-conv scatter ----------------
__global__ __launch_bounds__(256) void scatter_kernel(const float* __restrict__ Xmix,
                                                      const float* __restrict__ G,
                                                      float* __restrict__ out) {
  __shared__ float Gs[81];
  int p  = blockIdx.x;
  int wq = p % 30;
  int hq = (p / 30) % 30;
  int b  = p / 900;
  int o  = threadIdx.x;

  if (o < 81) Gs[o] = G[o];
  __syncthreads();

  float acc[9];
#pragma unroll
  for (int e = 0; e < 9; ++e) acc[e] = 0.f;

#pragma unroll
  for (int a = 0; a < 3; ++a) {
    int h = hq - a;
    if (h < 0 || h >= H_) continue;
#pragma unroll
    for (int bb = 0; bb < 3; ++bb) {
      int w = wq - bb;
      if (w < 0 || w >= W_) continue;
      const float* Gk = &Gs[(a * 3 + bb) * 9];
      const float* src = Xmix + ((size_t)b * M_B + (h * W_ + w) * 9) * CO_ + o;
      float Xm[9];
#pragma unroll
      for (int m = 0; m < 9; ++m) Xm[m] = src[(size_t)m * CO_];
      float T[9];
#pragma unroll
      for (int i = 0; i < 3; ++i)
#pragma unroll
        for (int l = 0; l < 3; ++l)
          T[i * 3 + l] = Gk[i * 3 + 0] * Xm[0 * 3 + l]
                       + Gk[i * 3 + 1] * Xm[1 * 3 + l]
                       + Gk[i * 3 + 2] * Xm[2 * 3 + l];
#pragma unroll
      for (int i = 0; i < 3; ++i)
#pragma unroll
        for (int mr = 0; mr < 3; ++mr)
          acc[i * 3 + mr] += T[i * 3 + 0] * Gk[mr * 3 + 0]
                           + T[i * 3 + 1] * Gk[mr * 3 + 1]
                           + T[i * 3 + 2] * Gk[mr * 3 + 2];
    }
  }

  float* dst = out + (size_t)(b * 900 + hq * 30 + wq) * 9 * CO_ + o;
#pragma unroll
  for (int e = 0; e < 9; ++e) dst[(size_t)e * CO_] = acc[e];
}

extern "C" void kernel_launch(void* const* d_in, const int* in_sizes, int n_in,
                              void* d_out, int out_size, void* d_ws, size_t ws_size,
                              hipStream_t stream) {
  const float* x = (const float*)d_in[0];   // (8,256,28,28,9) f32
  const float* g = (const float*)d_in[1];   // (24 + 256*256) f32
  float* out = (float*)d_out;               // (8,30,30,3,3,256) f32

  float* G    = (float*)d_ws;                      // 81 floats (padded to 128)
  float* csq  = (float*)d_ws + 128;                // 65536 floats
  float* Xmix = (float*)d_ws + 128 + C_ * CO_;     // 8*7056*256 floats = 57.8 MB

  cayley_kernel<<<1, 32, 0, stream>>>(g, G);
  sq_kernel<<<(C_ * CO_) / 256, 256, 0, stream>>>(g, csq);
  mix_gemm_kernel<<<B_ * MTILES, 256, LDS_DW * 4, stream>>>(x, csq, Xmix);
  scatter_kernel<<<B_ * 30 * 30, 256, 0, stream>>>(Xmix, G, out);
}